// BaseEBM_54717883351297
// MI455X (gfx1250) — compile-verified
//
#include <hip/hip_runtime.h>
#include <math.h>

typedef float v2f __attribute__((ext_vector_type(2)));
typedef float v8f __attribute__((ext_vector_type(8)));

#define STEPS   20
#define WIDTH   48
#define LRATE   0.1f
#define BETA1   0.9f
#define BETA2   0.999f
#define AEPS    1e-8f
#define HSTRIDE 52            // padded row stride (floats) for LDS staging
#define WAVES   4             // waves per block; 16 samples per wave

static __device__ __forceinline__ v8f wmma_f32_k4(v2f a, v2f b, v8f c) {
    // D = A(16x4,f32) * B(4x16,f32) + C(16x16,f32)  -> v_wmma_f32_16x16x4_f32
    return __builtin_amdgcn_wmma_f32_16x16x4_f32(false, a, false, b, (short)0, c,
                                                 false, false);
}

// Wave-local LDS fence: all staging buffers are wave-private, so cross-lane
// RAW/WAR inside the wave only needs (1) a compiler scheduling fence and
// (2) completion of outstanding DS ops (s_wait_dscnt 0). LDS ops from one
// wave execute in order, so no block-wide s_barrier rendezvous is required.
static __device__ __forceinline__ void wave_lds_fence() {
    __builtin_amdgcn_wave_barrier();
    asm volatile("s_wait_dscnt 0x0" ::: "memory");
    __builtin_amdgcn_wave_barrier();
}

// swish f(v)=v*sigmoid(v); propagate (value, d1, d2) truncated Taylor chain
static __device__ __forceinline__ void swish_chain(float pv, float pd, float ps,
                                                   float& ov, float& od, float& os) {
    float s  = 1.0f / (1.0f + __expf(-pv));
    float f1 = s + pv * s * (1.0f - s);                          // f'
    float f2 = s * (1.0f - s) * (2.0f + pv * (1.0f - 2.0f * s)); // f''
    ov = pv * s;
    od = f1 * pd;
    os = f2 * pd * pd + f1 * ps;
}

__global__ __launch_bounds__(32 * WAVES, 1)
void ebm_taylor_adam_kernel(const float* __restrict__ x,  const float* __restrict__ eps,
                            const float* __restrict__ W1, const float* __restrict__ b1,
                            const float* __restrict__ W2, const float* __restrict__ b2,
                            const float* __restrict__ W3, const float* __restrict__ b3,
                            const float* __restrict__ W4, const float* __restrict__ b4,
                            float* __restrict__ y_out, int Btot)
{
    (void)b4; // b4 does not affect dE/dy
    __shared__ float sW2[WIDTH * WIDTH];
    __shared__ float sW3[WIDTH * WIDTH];
    __shared__ float sW1[WIDTH * 2];
    __shared__ float sb1[WIDTH], sb2[WIDTH], sb3[WIDTH], sW4[WIDTH];
    __shared__ float stage[WAVES][3][16 * HSTRIDE];   // hv / hd / hs per wave
    __shared__ float sy0[WAVES][16];
    __shared__ float sx [WAVES][16];

    const int tid  = threadIdx.x;
    const int wave = tid >> 5;
    const int lane = tid & 31;

    for (int i = tid; i < WIDTH * WIDTH; i += blockDim.x) { sW2[i] = W2[i]; sW3[i] = W3[i]; }
    for (int i = tid; i < WIDTH * 2;     i += blockDim.x) { sW1[i] = W1[i]; }
    for (int i = tid; i < WIDTH;         i += blockDim.x) {
        sb1[i] = b1[i]; sb2[i] = b2[i]; sb3[i] = b3[i]; sW4[i] = W4[i];
    }

    const int sample_base = blockIdx.x * (WAVES * 16) + wave * 16;
    const int msel = lane & 15;
    const float xm = x[sample_base + msel];
    float y = 0.0f, mAdam = 0.0f, vAdam = 0.0f;   // Adam state, valid in lanes 0..15
    float b1pow = 1.0f, b2pow = 1.0f;
    if (lane < 16) sx[wave][lane] = xm;
    __syncthreads();   // the ONLY block-wide barrier: weights are shared by all 4 waves

    float* hv = stage[wave][0];
    float* hd = stage[wave][1];
    float* hs = stage[wave][2];

    const float* Wlayer[2] = { sW2, sW3 };
    const float* blayer[2] = { sb2, sb3 };

    for (int t = 0; t < STEPS; ++t) {
        const float e  = eps[t * Btot + sample_base + msel];
        const float y0 = y + e;                    // Taylor anchor
        if (lane < 16) sy0[wave][lane] = y0;
        wave_lds_fence();

        // ---- layer 1 (2 -> 48): VALU, write staging in sample-major layout ----
        for (int jj = 0; jj < 3; ++jj) {
            const int j = (lane & 15) + 16 * jj;
            const float w0 = sW1[j * 2 + 0], w1 = sW1[j * 2 + 1], bb = sb1[j];
            for (int i = 0; i < 8; ++i) {
                const int m = (lane >> 4) + 2 * i;
                const float pv = w0 * sx[wave][m] + w1 * sy0[wave][m] + bb;
                float ov, od, os;
                swish_chain(pv, w1, 0.0f, ov, od, os);
                hv[m * HSTRIDE + j] = ov;
                hd[m * HSTRIDE + j] = od;
                hs[m * HSTRIDE + j] = os;
            }
        }
        wave_lds_fence();

        // ---- layers 2,3 (48 -> 48): V_WMMA_F32_16X16X4_F32 ----
        for (int L = 0; L < 2; ++L) {
            const float* W  = Wlayer[L];
            const float* bb = blayer[L];
            v8f acc[3][3];
            for (int q = 0; q < 3; ++q)
                for (int nt = 0; nt < 3; ++nt) acc[q][nt] = {};

            // A fragment: lane<16 holds (M=lane, K=k0,k0+1); lane>=16 (M=lane-16, K=k0+2,k0+3)
            const int am  = lane & 15;
            const int akb = 2 * (lane >> 4);
            const int bn  = lane & 15;
            for (int kt = 0; kt < 12; ++kt) {
                const int ak = kt * 4 + akb;
                v2f aV, aD, aS;
                aV.x = hv[am * HSTRIDE + ak]; aV.y = hv[am * HSTRIDE + ak + 1];
                aD.x = hd[am * HSTRIDE + ak]; aD.y = hd[am * HSTRIDE + ak + 1];
                aS.x = hs[am * HSTRIDE + ak]; aS.y = hs[am * HSTRIDE + ak + 1];
                for (int nt = 0; nt < 3; ++nt) {
                    const int n = nt * 16 + bn;
                    // B[K][N] = W[N][K] (out = in @ W^T); mirror A's K-split over lane halves
                    v2f bF;
                    bF.x = W[n * WIDTH + ak];
                    bF.y = W[n * WIDTH + ak + 1];
                    acc[0][nt] = wmma_f32_k4(aV, bF, acc[0][nt]);   // value
                    acc[1][nt] = wmma_f32_k4(aD, bF, acc[1][nt]);   // d/dy
                    acc[2][nt] = wmma_f32_k4(aS, bF, acc[2][nt]);   // d2/dy2
                }
            }
            wave_lds_fence();   // staging reads done -> safe to overwrite

            // C/D layout: VGPR r, lane l -> (M = r + 8*(l>=16), N = l%16)
            const int mb = 8 * (lane >> 4);
            for (int nt = 0; nt < 3; ++nt) {
                const int n = nt * 16 + (lane & 15);
                const float bias = bb[n];
                for (int r = 0; r < 8; ++r) {
                    float ov, od, os;
                    swish_chain(acc[0][nt][r] + bias, acc[1][nt][r], acc[2][nt][r],
                                ov, od, os);
                    hv[(mb + r) * HSTRIDE + n] = ov;
                    hd[(mb + r) * HSTRIDE + n] = od;
                    hs[(mb + r) * HSTRIDE + n] = os;
                }
            }
            wave_lds_fence();
        }

        // ---- layer 4 (48 -> 1): lanes 0-15 reduce E'(y0), lanes 16-31 reduce E''(y0) ----
        const float* src = (lane < 16) ? hd : hs;
        float acc4 = 0.0f;
        for (int j = 0; j < WIDTH; ++j) acc4 += sW4[j] * src[msel * HSTRIDE + j];
        const float Ep  = acc4;
        const float Epp = __shfl(acc4, lane + 16, 32);  // valid in lanes 0..15

        // g = E'(y0) + E''(y0) * (y - y0),  y - y0 == -eps
        const float g = Ep + Epp * (-e);
        b1pow *= BETA1; b2pow *= BETA2;
        mAdam = BETA1 * mAdam + (1.0f - BETA1) * g;
        vAdam = BETA2 * vAdam + (1.0f - BETA2) * g * g;
        const float mhat = mAdam / (1.0f - b1pow);
        const float vhat = vAdam / (1.0f - b2pow);
        if (lane < 16) y = y - LRATE * mhat / (sqrtf(vhat) + AEPS);
        wave_lds_fence();   // layer-4 reads done before next step overwrites staging
    }

    if (lane < 16) y_out[sample_base + lane] = y;
}

extern "C" void kernel_launch(void* const* d_in, const int* in_sizes, int n_in,
                              void* d_out, int out_size, void* d_ws, size_t ws_size,
                              hipStream_t stream) {
    (void)n_in; (void)out_size; (void)d_ws; (void)ws_size;
    const float* x   = (const float*)d_in[0];
    const float* eps = (const float*)d_in[1];
    const float* W1  = (const float*)d_in[2];
    const float* b1  = (const float*)d_in[3];
    const float* W2  = (const float*)d_in[4];
    const float* b2  = (const float*)d_in[5];
    const float* W3  = (const float*)d_in[6];
    const float* b3  = (const float*)d_in[7];
    const float* W4  = (const float*)d_in[8];
    const float* b4  = (const float*)d_in[9];
    float* out = (float*)d_out;

    const int Btot   = in_sizes[0];         // 32768
    const int blocks = Btot / (WAVES * 16); // 64 samples per block
    ebm_taylor_adam_kernel<<<blocks, 32 * WAVES, 0, stream>>>(
        x, eps, W1, b1, W2, b2, W3, b3, W4, b4, out, Btot);
}